// GATWithEdgeAttr_52639119180343
// MI455X (gfx1250) — compile-verified
//
#include <hip/hip_runtime.h>
#include <hip/hip_bf16.h>
#include <limits.h>

// ---------------------------------------------------------------------------
// GAT with edge attributes, 2 layers + global mean pool.  MI455X / gfx1250.
// Dense GEMMs use V_WMMA_F32_16X16X4_F32 (full f32 precision matrix cores);
// edge phase is scatter/gather + atomics (the memory-bound part).
// ---------------------------------------------------------------------------

#define F_IN   128
#define ED     16
#define H1     8
#define C1     8
#define F1     (H1 * C1)   // 64
#define OUT    128
#define NGRAPH 256
#define NEG_SLOPE 0.2f

typedef __attribute__((ext_vector_type(2))) float v2f;
typedef __attribute__((ext_vector_type(8))) float v8f;

static __host__ __device__ inline int ceil_div(long long a, long long b) {
    return (int)((a + b - 1) / b);
}

// ------------------------- float <-> ordered int ---------------------------
__device__ __forceinline__ int f2ord(float f) {
    int i = __float_as_int(f);
    return (i >= 0) ? i : (i ^ 0x7FFFFFFF);
}
__device__ __forceinline__ float ord2f(int k) {
    int i = (k >= 0) ? k : (k ^ 0x7FFFFFFF);
    return __int_as_float(i);
}

// ------------------------------ WMMA GEMM ----------------------------------
// C[M,Ncols] = A[M,K] @ B[K,Ncols], all f32 row-major.
// One wave (32 lanes) per 16x16 output tile, K stepped by 4 using
// V_WMMA_F32_16X16X4_F32.
//  A 16x4 layout: lanes 0-15 -> rows M=0..15 with K=kb+0,kb+1 in the 2 VGPRs;
//                 lanes 16-31 -> same rows with K=kb+2,kb+3.
//  B 4x16 layout: mirrored (lanes index columns N, halves index K pairs).
//  C/D 16x16:     VGPR v holds M = v (lanes 0-15) / M = v+8 (lanes 16-31),
//                 N = lane & 15.
__global__ void gemm_f32_wmma(const float* __restrict__ A,
                              const float* __restrict__ B,
                              float* __restrict__ C,
                              int M, int K, int Ncols) {
    const int lane = threadIdx.x;          // 0..31 (one wave per block)
    const int half = lane >> 4;            // 0 or 1
    const int l15  = lane & 15;
    const int colBase = blockIdx.x * 16;
    const int rowBase = blockIdx.y * 16;

    const int mRow = rowBase + l15;
    const int mCl  = (mRow < M) ? mRow : (M - 1);   // clamp; keep EXEC full
    const int nCol = colBase + l15;

    v8f acc = {};
    for (int kb = 0; kb < K; kb += 4) {
        const int k0 = kb + half * 2;
        v2f a, b;
        a.x = A[(size_t)mCl * K + k0];
        a.y = A[(size_t)mCl * K + k0 + 1];
        b.x = B[(size_t)k0 * Ncols + nCol];
        b.y = B[(size_t)(k0 + 1) * Ncols + nCol];
        acc = __builtin_amdgcn_wmma_f32_16x16x4_f32(
            /*neg_a=*/false, a, /*neg_b=*/false, b,
            /*c_mod=*/(short)0, acc, /*reuse_a=*/false, /*reuse_b=*/false);
    }

    const int mOut = rowBase + half * 8;
#pragma unroll
    for (int v = 0; v < 8; ++v) {
        const int m = mOut + v;
        if (m < M) C[(size_t)m * Ncols + nCol] = acc[v];
    }
}

// ------------------------- init / small kernels ----------------------------
__global__ void fill_i32(int* p, int v, long long n) {
    long long i = (long long)blockIdx.x * blockDim.x + threadIdx.x;
    if (i < n) p[i] = v;
}

// deg[dst] += 1 ; easum[dst][:] += edge_attr[e][:]
__global__ void deg_easum_kernel(const int* __restrict__ src,
                                 const int* __restrict__ dst,
                                 const float* __restrict__ ea,
                                 int* __restrict__ deg,
                                 float* __restrict__ easum, int E) {
    int e = blockIdx.x * blockDim.x + threadIdx.x;
    if (e >= E) return;
    int d = dst[e];
    atomicAdd(&deg[d], 1);
#pragma unroll
    for (int j = 0; j < ED; ++j)
        atomicAdd(&easum[(size_t)d * ED + j], ea[(size_t)e * ED + j]);
}

// easum /= max(deg,1)   (in place -> ea_self)
__global__ void eaself_norm_kernel(float* __restrict__ easum,
                                   const int* __restrict__ deg, int n) {
    long long i = (long long)blockIdx.x * blockDim.x + threadIdx.x;
    if (i >= (long long)n * ED) return;
    int node = (int)(i / ED);
    float dg = (float)deg[node];
    dg = dg > 1.0f ? dg : 1.0f;
    easum[i] /= dg;
}

// al_s[n,h] = sum_c xw[n,h,c]*att_s[h,c] ; same for al_d
__global__ void node_scores_kernel(const float* __restrict__ xw,
                                   const float* __restrict__ att_s,
                                   const float* __restrict__ att_d,
                                   float* __restrict__ al_s,
                                   float* __restrict__ al_d,
                                   int n, int heads, int ch) {
    int i = blockIdx.x * blockDim.x + threadIdx.x;
    if (i >= n * heads) return;
    int node = i / heads, h = i % heads;
    const float* row = xw + (size_t)node * heads * ch + (size_t)h * ch;
    float ss = 0.f, sd = 0.f;
    for (int c = 0; c < ch; ++c) {
        float v = row[c];
        ss += v * att_s[h * ch + c];
        sd += v * att_d[h * ch + c];
    }
    al_s[i] = ss;
    al_d[i] = sd;
}

// Ve[k,h] = sum_c We[k, h*ch + c] * att_e[h,c]    (ED x heads)
__global__ void make_ve_kernel(const float* __restrict__ We,
                               const float* __restrict__ att_e,
                               float* __restrict__ Ve, int heads, int ch) {
    int i = blockIdx.x * blockDim.x + threadIdx.x;
    if (i >= ED * heads) return;
    int k = i / heads, h = i % heads;
    float s = 0.f;
    for (int c = 0; c < ch; ++c)
        s += We[(size_t)k * heads * ch + (size_t)h * ch + c] * att_e[h * ch + c];
    Ve[k * heads + h] = s;
}

// raw attention logits + running segment max (ordered-int atomicMax).
// Edges e < E are real; e >= E are self loops of node (e-E) using ea_self.
__global__ void alpha_raw_kernel(const int* __restrict__ src,
                                 const int* __restrict__ dst,
                                 const float* __restrict__ ea,
                                 const float* __restrict__ ea_self,
                                 const float* __restrict__ Ve,
                                 const float* __restrict__ al_s,
                                 const float* __restrict__ al_d,
                                 float* __restrict__ alpha,
                                 int* __restrict__ nmax,
                                 int E, int n, int heads) {
    int e = blockIdx.x * blockDim.x + threadIdx.x;
    int EP = E + n;
    if (e >= EP) return;
    int s, d;
    const float* eap;
    if (e < E) { s = src[e]; d = dst[e]; eap = ea + (size_t)e * ED; }
    else       { s = d = e - E;          eap = ea_self + (size_t)(e - E) * ED; }
    float eav[ED];
#pragma unroll
    for (int j = 0; j < ED; ++j) eav[j] = eap[j];
    for (int h = 0; h < heads; ++h) {
        float ale = 0.f;
#pragma unroll
        for (int j = 0; j < ED; ++j) ale += eav[j] * Ve[j * heads + h];
        float a = al_s[(size_t)s * heads + h] + al_d[(size_t)d * heads + h] + ale;
        a = (a > 0.f) ? a : NEG_SLOPE * a;   // leaky relu
        alpha[(size_t)e * heads + h] = a;
        atomicMax(&nmax[(size_t)d * heads + h], f2ord(a));
    }
}

// alpha <- exp(alpha - max[dst]); nsum[dst] += alpha
__global__ void alpha_exp_kernel(const int* __restrict__ dst,
                                 float* __restrict__ alpha,
                                 const int* __restrict__ nmax,
                                 float* __restrict__ nsum,
                                 int E, int n, int heads) {
    long long idx = (long long)blockIdx.x * blockDim.x + threadIdx.x;
    long long total = (long long)(E + n) * heads;
    if (idx >= total) return;
    int e = (int)(idx / heads);
    int h = (int)(idx % heads);
    int d = (e < E) ? dst[e] : (e - E);
    float m  = ord2f(nmax[(size_t)d * heads + h]);
    float ex = expf(alpha[idx] - m);
    alpha[idx] = ex;
    atomicAdd(&nsum[(size_t)d * heads + h], ex);
}

// out[dst, c] += xw[src, c] * alpha[e,h]/(nsum[dst,h]+eps)
__global__ void aggregate_kernel(const int* __restrict__ src,
                                 const int* __restrict__ dst,
                                 const float* __restrict__ xw,
                                 const float* __restrict__ alpha,
                                 const float* __restrict__ nsum,
                                 float* __restrict__ out,
                                 int E, int n, int heads, int ch) {
    const int F = heads * ch;
    long long idx = (long long)blockIdx.x * blockDim.x + threadIdx.x;
    long long total = (long long)(E + n) * F;
    if (idx >= total) return;
    int e = (int)(idx / F);
    int c = (int)(idx % F);
    int h = c / ch;
    int s, d;
    if (e < E) { s = src[e]; d = dst[e]; } else { s = d = e - E; }
    float w = alpha[(size_t)e * heads + h] /
              (nsum[(size_t)d * heads + h] + 1e-16f);
    atomicAdd(&out[(size_t)d * F + c], xw[(size_t)s * F + c] * w);
}

// h <- h + b  (optionally ELU)
__global__ void bias_act_kernel(float* __restrict__ h,
                                const float* __restrict__ b,
                                long long n, int F, int do_elu) {
    long long idx = (long long)blockIdx.x * blockDim.x + threadIdx.x;
    if (idx >= n * F) return;
    int c = (int)(idx % F);
    float v = h[idx] + b[c];
    if (do_elu) v = (v > 0.f) ? v : (expf(v) - 1.0f);
    h[idx] = v;
}

// pooled-sum + per-graph counts
__global__ void pool_sum_kernel(const float* __restrict__ h,
                                const int* __restrict__ batch,
                                float* __restrict__ psum,
                                int* __restrict__ cnt, int n, int F) {
    long long idx = (long long)blockIdx.x * blockDim.x + threadIdx.x;
    if (idx >= (long long)n * F) return;
    int node = (int)(idx / F);
    int c    = (int)(idx % F);
    int g = batch[node];
    atomicAdd(&psum[(size_t)g * F + c], h[idx]);
    if (c == 0) atomicAdd(&cnt[g], 1);
}

__global__ void pool_div_kernel(const float* __restrict__ psum,
                                const int* __restrict__ cnt,
                                float* __restrict__ out, int F) {
    long long idx = (long long)blockIdx.x * blockDim.x + threadIdx.x;
    if (idx >= (long long)NGRAPH * F) return;
    int g = (int)(idx / F);
    float c = (float)cnt[g];
    c = c > 1.f ? c : 1.f;
    out[idx] = psum[idx] / c;
}

// ---------------------------------------------------------------------------
extern "C" void kernel_launch(void* const* d_in, const int* in_sizes, int n_in,
                              void* d_out, int out_size, void* d_ws, size_t ws_size,
                              hipStream_t stream) {
    const float* x    = (const float*)d_in[0];
    const int*   eidx = (const int*)d_in[1];
    const float* ea   = (const float*)d_in[2];
    const int*   batch= (const int*)d_in[3];
    const float* W1   = (const float*)d_in[4];
    const float* as1  = (const float*)d_in[5];
    const float* ad1  = (const float*)d_in[6];
    const float* ae1  = (const float*)d_in[7];
    const float* We1  = (const float*)d_in[8];
    const float* b1   = (const float*)d_in[9];
    const float* W2   = (const float*)d_in[10];
    const float* as2  = (const float*)d_in[11];
    const float* ad2  = (const float*)d_in[12];
    const float* ae2  = (const float*)d_in[13];
    const float* We2  = (const float*)d_in[14];
    const float* b2   = (const float*)d_in[15];
    float* outp = (float*)d_out;

    const int N  = in_sizes[0] / F_IN;
    const int E  = in_sizes[1] / 2;
    const int EP = E + N;
    const int* src = eidx;
    const int* dst = eidx + E;

    // ------------------- workspace carve-out (256B aligned) ----------------
    size_t off = 0;
    auto take = [&](size_t bytes) -> void* {
        void* p = (char*)d_ws + off;
        off += (bytes + 255) & ~(size_t)255;
        return p;
    };
    int*   deg    = (int*)  take((size_t)N * 4);
    float* eaself = (float*)take((size_t)N * ED * 4);
    float* xw     = (float*)take((size_t)N * F1 * 4);
    float* al_s1  = (float*)take((size_t)N * H1 * 4);
    float* al_d1  = (float*)take((size_t)N * H1 * 4);
    float* Ve1    = (float*)take((size_t)ED * H1 * 4);
    float* alpha1 = (float*)take((size_t)EP * H1 * 4);
    int*   nmax1  = (int*)  take((size_t)N * H1 * 4);
    float* nsum1  = (float*)take((size_t)N * H1 * 4);
    float* agg1   = (float*)take((size_t)N * F1 * 4);   // becomes h1
    float* hw     = (float*)take((size_t)N * OUT * 4);
    float* al_s2  = (float*)take((size_t)N * 4);
    float* al_d2  = (float*)take((size_t)N * 4);
    float* Ve2    = (float*)take((size_t)ED * 4);
    float* alpha2 = (float*)take((size_t)EP * 4);
    int*   nmax2  = (int*)  take((size_t)N * 4);
    float* nsum2  = (float*)take((size_t)N * 4);
    float* agg2   = (float*)take((size_t)N * OUT * 4);  // becomes h2
    float* psum   = (float*)take((size_t)NGRAPH * OUT * 4);
    int*   cnt    = (int*)  take((size_t)NGRAPH * 4);
    (void)ws_size; (void)n_in; (void)out_size;

    const int T = 256;

    // ------------------------------- init ----------------------------------
    hipMemsetAsync(deg,    0, (size_t)N * 4, stream);
    hipMemsetAsync(eaself, 0, (size_t)N * ED * 4, stream);
    hipMemsetAsync(nsum1,  0, (size_t)N * H1 * 4, stream);
    hipMemsetAsync(agg1,   0, (size_t)N * F1 * 4, stream);
    hipMemsetAsync(nsum2,  0, (size_t)N * 4, stream);
    hipMemsetAsync(agg2,   0, (size_t)N * OUT * 4, stream);
    hipMemsetAsync(psum,   0, (size_t)NGRAPH * OUT * 4, stream);
    hipMemsetAsync(cnt,    0, (size_t)NGRAPH * 4, stream);
    fill_i32<<<ceil_div((long long)N * H1, T), T, 0, stream>>>(nmax1, INT_MIN, (long long)N * H1);
    fill_i32<<<ceil_div(N, T), T, 0, stream>>>(nmax2, INT_MIN, N);

    // -------------------- self-loop edge attr (fill='mean') ----------------
    deg_easum_kernel<<<ceil_div(E, T), T, 0, stream>>>(src, dst, ea, deg, eaself, E);
    eaself_norm_kernel<<<ceil_div((long long)N * ED, T), T, 0, stream>>>(eaself, deg, N);

    // ------------------------------ layer 1 --------------------------------
    {
        dim3 g(F1 / 16, ceil_div(N, 16));
        gemm_f32_wmma<<<g, 32, 0, stream>>>(x, W1, xw, N, F_IN, F1);
    }
    node_scores_kernel<<<ceil_div(N * H1, T), T, 0, stream>>>(xw, as1, ad1, al_s1, al_d1, N, H1, C1);
    make_ve_kernel<<<1, ED * H1, 0, stream>>>(We1, ae1, Ve1, H1, C1);
    alpha_raw_kernel<<<ceil_div(EP, T), T, 0, stream>>>(src, dst, ea, eaself, Ve1,
                                                        al_s1, al_d1, alpha1, nmax1, E, N, H1);
    alpha_exp_kernel<<<ceil_div((long long)EP * H1, T), T, 0, stream>>>(dst, alpha1, nmax1, nsum1, E, N, H1);
    aggregate_kernel<<<ceil_div((long long)EP * F1, T), T, 0, stream>>>(src, dst, xw, alpha1, nsum1,
                                                                        agg1, E, N, H1, C1);
    bias_act_kernel<<<ceil_div((long long)N * F1, T), T, 0, stream>>>(agg1, b1, N, F1, /*elu=*/1);

    // ------------------------------ layer 2 --------------------------------
    {
        dim3 g(OUT / 16, ceil_div(N, 16));
        gemm_f32_wmma<<<g, 32, 0, stream>>>(agg1, W2, hw, N, F1, OUT);
    }
    node_scores_kernel<<<ceil_div(N, T), T, 0, stream>>>(hw, as2, ad2, al_s2, al_d2, N, 1, OUT);
    make_ve_kernel<<<1, ED, 0, stream>>>(We2, ae2, Ve2, 1, OUT);
    alpha_raw_kernel<<<ceil_div(EP, T), T, 0, stream>>>(src, dst, ea, eaself, Ve2,
                                                        al_s2, al_d2, alpha2, nmax2, E, N, 1);
    alpha_exp_kernel<<<ceil_div(EP, T), T, 0, stream>>>(dst, alpha2, nmax2, nsum2, E, N, 1);
    aggregate_kernel<<<ceil_div((long long)EP * OUT, T), T, 0, stream>>>(src, dst, hw, alpha2, nsum2,
                                                                         agg2, E, N, 1, OUT);
    bias_act_kernel<<<ceil_div((long long)N * OUT, T), T, 0, stream>>>(agg2, b2, N, OUT, /*elu=*/0);

    // ------------------------- global mean pool ----------------------------
    pool_sum_kernel<<<ceil_div((long long)N * OUT, T), T, 0, stream>>>(agg2, batch, psum, cnt, N, OUT);
    pool_div_kernel<<<ceil_div((long long)NGRAPH * OUT, T), T, 0, stream>>>(psum, cnt, outp, OUT);
}